// Unet3DCrossBiGeneric_57346403336678
// MI455X (gfx1250) — compile-verified
//
#include <hip/hip_runtime.h>
#include <hip/hip_bf16.h>
#include <math.h>

// ---------------- problem constants ----------------
#define BB   2
#define CC   128
#define DD   16
#define HH   16
#define WW   16
#define NN   (DD*HH*WW)          // 4096
#define EE   (DD+HH+WW-2)        // 46
#define DSIM 64
#define GDIM 64
#define PSIM 64
#define BN   (BB*NN)             // 8192
#define PROJW 192                // theta(64) | phi(64) | g(64)
#define NROUNDS 3

typedef __attribute__((ext_vector_type(2))) float v2f;
typedef __attribute__((ext_vector_type(8))) float v8f;

// Sorted cross-graph neighbor index, closed form (replicates np.sort of the
// concatenated axis lines, self kept once).
__device__ __forceinline__ int nbr_idx(int i, int j, int k, int e) {
    if (e < i)          return e * 256 + j * 16 + k;            // slabs below
    if (e < i + j)      return i * 256 + (e - i) * 16 + k;      // j' < j
    if (e < i + j + 16) return i * 256 + j * 16 + (e - i - j);  // full row (incl self)
    if (e < i + 31)     return i * 256 + (e - i - 15) * 16 + k; // j' > j
    return (e - 30) * 256 + j * 16 + k;                         // slabs above
}

// ---------------- WMMA GEMM: fused projections ----------------
// proj[m, 0:64]=theta, [64:128]=phi, [128:192]=g ;  proj = h @ Wcat^T + bcat
// grid: (BN/16, 192/16), block: 32 (one wave per 16x16 tile)
__global__ void proj_gemm_kernel(const float* __restrict__ h,
                                 const float* __restrict__ theta_w,
                                 const float* __restrict__ theta_b,
                                 const float* __restrict__ phi_w,
                                 const float* __restrict__ phi_b,
                                 const float* __restrict__ G_w,
                                 const float* __restrict__ G_b,
                                 float* __restrict__ proj) {
    const int m0   = blockIdx.x * 16;
    const int n0   = blockIdx.y * 16;
    const int lane = threadIdx.x;
    const int half = lane >> 4;     // 0: K={0,1}, 1: K={2,3}
    const int lid  = lane & 15;     // A-row / B-col / D-col within tile

    const int ncol = n0 + lid;      // 0..191
    const float* Wm; const float* bias; int wc;
    if (ncol < 64)       { Wm = theta_w; bias = theta_b; wc = ncol; }
    else if (ncol < 128) { Wm = phi_w;   bias = phi_b;   wc = ncol - 64; }
    else                 { Wm = G_w;     bias = G_b;     wc = ncol - 128; }

    const float* arow = h  + (size_t)(m0 + lid) * CC;  // A[M=lid, :]
    const float* wrow = Wm + (size_t)wc * CC;          // B[k, N=lid] = W[N, k]

    v8f acc = {};
    #pragma unroll
    for (int k = 0; k < CC; k += 4) {
        v2f a = *(const v2f*)(arow + k + 2 * half);
        v2f b = *(const v2f*)(wrow + k + 2 * half);
        acc = __builtin_amdgcn_wmma_f32_16x16x4_f32(
            false, a, false, b, (short)0, acc, false, false);
    }
    const float bv = bias[wc];
    #pragma unroll
    for (int v = 0; v < 8; ++v) {
        const int m = m0 + v + 8 * half;               // D: M = v (+8 for hi half)
        proj[(size_t)m * PROJW + n0 + lid] = acc[v] + bv;
    }
}

// ---------------- WMMA GEMM: residual  h += y @ r_w^T + r_b --------------
// grid: (BN/16, 128/16), block: 32
__global__ void residual_gemm_kernel(const float* __restrict__ y,
                                     const float* __restrict__ r_w,
                                     const float* __restrict__ r_b,
                                     float* __restrict__ h) {
    const int m0   = blockIdx.x * 16;
    const int n0   = blockIdx.y * 16;
    const int lane = threadIdx.x;
    const int half = lane >> 4;
    const int lid  = lane & 15;

    const float* arow = y   + (size_t)(m0 + lid) * GDIM;
    const float* wrow = r_w + (size_t)(n0 + lid) * GDIM;  // B[g, c] = r_w[c, g]

    v8f acc = {};
    #pragma unroll
    for (int k = 0; k < GDIM; k += 4) {
        v2f a = *(const v2f*)(arow + k + 2 * half);
        v2f b = *(const v2f*)(wrow + k + 2 * half);
        acc = __builtin_amdgcn_wmma_f32_16x16x4_f32(
            false, a, false, b, (short)0, acc, false, false);
    }
    const float bv = r_b[n0 + lid];
    #pragma unroll
    for (int v = 0; v < 8; ++v) {
        const int m = m0 + v + 8 * half;
        const int c = n0 + lid;
        h[(size_t)m * CC + c] += acc[v] + bv;
    }
}

// ---------------- f[b,n,e] = <theta[b,n], phi[b,nbr(n,e)]> ----------------
// grid: BN, block: 64
__global__ void attn_f_kernel(const float* __restrict__ proj,
                              float* __restrict__ f) {
    __shared__ float th[DSIM];
    const int bn = blockIdx.x;
    const int b  = bn >> 12;
    const int n  = bn & (NN - 1);
    const int i = n >> 8, j = (n >> 4) & 15, k = n & 15;
    const int t = threadIdx.x;
    th[t] = proj[(size_t)bn * PROJW + t];
    __syncthreads();
    if (t < EE) {
        const int nb = nbr_idx(i, j, k, t);
        const float* ph = proj + (size_t)(b * NN + nb) * PROJW + DSIM;
        float acc = 0.f;
        #pragma unroll 8
        for (int d = 0; d < DSIM; ++d) acc = fmaf(th[d], ph[d], acc);
        f[(size_t)bn * EE + t] = acc;
    }
}

// ---------------- column sum-of-squares over nodes (deterministic) --------
// grid: nB*EE, block: 256 ; f layout: [(b*NN + n)*EE + e]
__global__ void sumsq_kernel(const float* __restrict__ f,
                             float* __restrict__ sumsq) {
    __shared__ float red[256];
    const int be = blockIdx.x;
    const int b = be / EE, e = be % EE;
    float acc = 0.f;
    for (int n = threadIdx.x; n < NN; n += 256) {
        const float v = f[(size_t)(b * NN + n) * EE + e];
        acc = fmaf(v, v, acc);
    }
    red[threadIdx.x] = acc;
    __syncthreads();
    for (int s = 128; s > 0; s >>= 1) {
        if (threadIdx.x < s) red[threadIdx.x] += red[threadIdx.x + s];
        __syncthreads();
    }
    if (threadIdx.x == 0) sumsq[be] = red[0];
}

// ------- softmax over e of (f/norm + f_p)  and  y = attn @ g[nbr] ---------
// grid: BN, block: 64 (t = output channel g)
__global__ void attn_softmax_y_kernel(const float* __restrict__ proj,
                                      const float* __restrict__ f,
                                      const float* __restrict__ sumsq,
                                      const float* __restrict__ f_p,
                                      float* __restrict__ y) {
    __shared__ float s[EE];
    const int bn = blockIdx.x;
    const int b  = bn >> 12;
    const int n  = bn & (NN - 1);
    const int i = n >> 8, j = (n >> 4) & 15, k = n & 15;
    const int t = threadIdx.x;
    if (t < EE) {
        const float nrm = 1e-6f + sqrtf(sumsq[b * EE + t]);
        s[t] = f[(size_t)bn * EE + t] / nrm + f_p[(size_t)n * EE + t];
    }
    __syncthreads();
    float mx = -INFINITY;
    #pragma unroll
    for (int e = 0; e < EE; ++e) mx = fmaxf(mx, s[e]);
    float sum = 0.f;
    #pragma unroll
    for (int e = 0; e < EE; ++e) sum += __expf(s[e] - mx);
    const float inv = 1.f / sum;
    float acc = 0.f;
    for (int e = 0; e < EE; ++e) {
        const float w  = __expf(s[e] - mx) * inv;
        const int   nb = nbr_idx(i, j, k, e);
        acc = fmaf(w, proj[(size_t)(b * NN + nb) * PROJW + 128 + t], acc);
    }
    y[(size_t)bn * GDIM + t] = acc;
}

// ---------------- geometric positional features ---------------------------
// grid: NN, block: 64
__global__ void geo_proj_kernel(const float* __restrict__ gtw,
                                const float* __restrict__ gtb,
                                const float* __restrict__ gpw,
                                const float* __restrict__ gpb,
                                float* __restrict__ pth,
                                float* __restrict__ pph) {
    const int n = blockIdx.x, d = threadIdx.x;
    const int i = n >> 8, j = (n >> 4) & 15, k = n & 15;
    const float p0 = (float)i / 15.f - 0.5f;
    const float p1 = (float)j / 15.f - 0.5f;
    const float p2 = (float)k / 15.f - 0.5f;
    pth[(size_t)n * PSIM + d] =
        fmaf(p0, gtw[d*3+0], fmaf(p1, gtw[d*3+1], fmaf(p2, gtw[d*3+2], gtb[d])));
    pph[(size_t)n * PSIM + d] =
        fmaf(p0, gpw[d*3+0], fmaf(p1, gpw[d*3+1], fmaf(p2, gpw[d*3+2], gpb[d])));
}

// f_p_raw[n,e] = <p_theta[n], p_phi[nbr]> ; grid: NN, block: 64
__global__ void geo_f_kernel(const float* __restrict__ pth,
                             const float* __restrict__ pph,
                             float* __restrict__ fp) {
    __shared__ float th[PSIM];
    const int n = blockIdx.x;
    const int i = n >> 8, j = (n >> 4) & 15, k = n & 15;
    const int t = threadIdx.x;
    th[t] = pth[(size_t)n * PSIM + t];
    __syncthreads();
    if (t < EE) {
        const int nb = nbr_idx(i, j, k, t);
        const float* ph = pph + (size_t)nb * PSIM;
        float acc = 0.f;
        #pragma unroll 8
        for (int d = 0; d < PSIM; ++d) acc = fmaf(th[d], ph[d], acc);
        fp[(size_t)n * EE + t] = acc;
    }
}

// in-place: f_p = relu(f_p_raw / (1e-6 + ||col||))
__global__ void geo_norm_kernel(float* __restrict__ fp,
                                const float* __restrict__ sumsq_p) {
    const int idx = blockIdx.x * blockDim.x + threadIdx.x;
    if (idx < NN * EE) {
        const int e = idx % EE;
        const float v = fp[idx] / (1e-6f + sqrtf(sumsq_p[e]));
        fp[idx] = fmaxf(v, 0.f);
    }
}

// ---------------- layout transposes ---------------------------------------
// h[b, n, c] = x[b, c, n]
__global__ void init_h_kernel(const float* __restrict__ x, float* __restrict__ h) {
    const int tid = blockIdx.x * blockDim.x + threadIdx.x;
    if (tid < BN * CC) {
        const int b = tid / (NN * CC);
        const int r = tid - b * NN * CC;
        const int n = r / CC;
        const int c = r - n * CC;
        h[tid] = x[(size_t)b * CC * NN + (size_t)c * NN + n];
    }
}
// out[b, c, n] = h[b, n, c]
__global__ void final_out_kernel(const float* __restrict__ h, float* __restrict__ out) {
    const int tid = blockIdx.x * blockDim.x + threadIdx.x;
    if (tid < BN * CC) {
        const int b = tid / (CC * NN);
        const int r = tid - b * CC * NN;
        const int c = r / NN;
        const int n = r - c * NN;
        out[tid] = h[(size_t)(b * NN + n) * CC + c];
    }
}

// ---------------- launch ---------------------------------------------------
extern "C" void kernel_launch(void* const* d_in, const int* in_sizes, int n_in,
                              void* d_out, int out_size, void* d_ws, size_t ws_size,
                              hipStream_t stream) {
    const float* x      = (const float*)d_in[0];
    const float* G_w    = (const float*)d_in[1];
    const float* G_b    = (const float*)d_in[2];
    const float* th_w   = (const float*)d_in[3];
    const float* th_b   = (const float*)d_in[4];
    const float* ph_w   = (const float*)d_in[5];
    const float* ph_b   = (const float*)d_in[6];
    const float* r_w    = (const float*)d_in[7];
    const float* r_b    = (const float*)d_in[8];
    const float* gth_w  = (const float*)d_in[9];
    const float* gth_b  = (const float*)d_in[10];
    const float* gph_w  = (const float*)d_in[11];
    const float* gph_b  = (const float*)d_in[12];
    float* out = (float*)d_out;

    float* ws      = (float*)d_ws;
    float* h       = ws;                         // BN*CC      = 1,048,576
    float* proj    = h + (size_t)BN * CC;        // BN*192     = 1,572,864
    float* f       = proj + (size_t)BN * PROJW;  // BN*EE      =   376,832
    float* y       = f + (size_t)BN * EE;        // BN*GDIM    =   524,288
    float* f_p     = y + (size_t)BN * GDIM;      // NN*EE      =   188,416
    float* pth     = f_p + (size_t)NN * EE;      // NN*PSIM    =   262,144
    float* pph     = pth + (size_t)NN * PSIM;    // NN*PSIM    =   262,144
    float* sumsq   = pph + (size_t)NN * PSIM;    // BB*EE (pad 128)
    float* sumsq_p = sumsq + 128;                // EE

    // --- geometric attention bias (once) ---
    geo_proj_kernel<<<NN, 64, 0, stream>>>(gth_w, gth_b, gph_w, gph_b, pth, pph);
    geo_f_kernel<<<NN, 64, 0, stream>>>(pth, pph, f_p);
    sumsq_kernel<<<EE, 256, 0, stream>>>(f_p, sumsq_p);      // b-dim == 1
    geo_norm_kernel<<<(NN * EE + 255) / 256, 256, 0, stream>>>(f_p, sumsq_p);

    // --- h = transpose(x) ---
    init_h_kernel<<<(BN * CC + 255) / 256, 256, 0, stream>>>(x, h);

    // --- 3 message-passing rounds ---
    for (int r = 0; r < NROUNDS; ++r) {
        proj_gemm_kernel<<<dim3(BN / 16, PROJW / 16), 32, 0, stream>>>(
            h, th_w, th_b, ph_w, ph_b, G_w, G_b, proj);
        attn_f_kernel<<<BN, 64, 0, stream>>>(proj, f);
        sumsq_kernel<<<BB * EE, 256, 0, stream>>>(f, sumsq);
        attn_softmax_y_kernel<<<BN, 64, 0, stream>>>(proj, f, sumsq, f_p, y);
        residual_gemm_kernel<<<dim3(BN / 16, CC / 16), 32, 0, stream>>>(y, r_w, r_b, h);
    }

    // --- output transpose ---
    final_out_kernel<<<(BN * CC + 255) / 256, 256, 0, stream>>>(h, out);

    (void)in_sizes; (void)n_in; (void)out_size; (void)ws_size;
}